// SimpleGCNNet_36283883716825
// MI455X (gfx1250) — compile-verified
//
#include <hip/hip_runtime.h>
#include <hip/hip_bf16.h>

typedef __attribute__((ext_vector_type(2))) float v2f;
typedef __attribute__((ext_vector_type(4))) float v4f;
typedef __attribute__((ext_vector_type(8))) float v8f;

#define Nn    64     // nodes
#define Ee    4096   // edges
#define Ff    2000   // features per node (Kk*T)
#define Hh    128    // output features
#define Bb    32     // batches
#define KB    40     // K-panel width
#define NP    (Ff / KB)      // 50 panels
#define LXS   44             // X panel LDS row stride (floats)
#define LWS   40             // W panel LDS row stride (floats)
#define PANEL (Nn * LXS + Hh * LWS)   // 7936 floats per buffer
#define XQ    (KB / 4)       // 10 float4 per row

#if defined(__gfx1250__)
#define HAVE_ASYNC 1
#else
#define HAVE_ASYNC 0
#endif

// ---------------------------------------------------------------------------
// Kernel 1: build the dense normalized 2-hop operator A^2 (64x64) from edges.
// ---------------------------------------------------------------------------
__global__ __launch_bounds__(256) void sgc_build_a2(
    const long long* __restrict__ ei,   // (2, E) int64: [src ; dst]
    const float* __restrict__ ew,       // (E,)
    float* __restrict__ A2out)          // (64*64,) workspace
{
    __shared__ float deg[Nn];
    __shared__ float dinv[Nn];
    __shared__ float A[Nn * Nn];

    const int tid = threadIdx.x;
    const long long* src = ei;
    const long long* dst = ei + Ee;

    for (int i = tid; i < Nn * Nn; i += 256) A[i] = 0.f;
    if (tid < Nn) deg[tid] = 0.f;
    __syncthreads();

    for (int e = tid; e < Ee; e += 256) {
        float w = ew[e];
        if (w <= 0.f) w = 1e-6f;
        atomicAdd(&deg[(int)dst[e]], w);
    }
    if (tid < Nn) atomicAdd(&deg[tid], 1.0f);   // self-loop weight 1.0
    __syncthreads();

    if (tid < Nn) {
        float d = deg[tid];
        dinv[tid] = (d > 0.f) ? rsqrtf(d) : 0.f;
    }
    __syncthreads();

    for (int e = tid; e < Ee; e += 256) {
        float w = ew[e];
        if (w <= 0.f) w = 1e-6f;
        int s = (int)src[e], d = (int)dst[e];
        atomicAdd(&A[d * Nn + s], dinv[s] * w * dinv[d]);
    }
    if (tid < Nn) atomicAdd(&A[tid * Nn + tid], dinv[tid] * dinv[tid]);
    __syncthreads();

    for (int i = tid; i < Nn * Nn; i += 256) {     // A2 = A @ A
        int m = i >> 6, n = i & 63;
        float s = 0.f;
        #pragma unroll 8
        for (int k = 0; k < Nn; ++k) s += A[m * Nn + k] * A[k * Nn + n];
        A2out[i] = s;
    }
}

// ---------------------------------------------------------------------------
// Async global->LDS staging of one K-panel (X: 64xKB, W: 128xKB) into buffer.
// 1920 float4 copies spread over 256 threads; tracked by ASYNCcnt.
// ---------------------------------------------------------------------------
__device__ __forceinline__ void stage_panel(
    const float* __restrict__ xb, const float* __restrict__ W,
    float* buff, unsigned bufOffBytes, int k0, int tid)
{
    for (int i = tid; i < (Nn + Hh) * XQ; i += 256) {
        const float* gsrc;
        unsigned loff;
        if (i < Nn * XQ) {
            int row = i / XQ, c4 = i % XQ;
            gsrc = xb + row * Ff + k0 + c4 * 4;
            loff = bufOffBytes + (unsigned)(row * LXS + c4 * 4) * 4u;
        } else {
            int j = i - Nn * XQ;
            int row = j / XQ, c4 = j % XQ;
            gsrc = W + row * Ff + k0 + c4 * 4;
            loff = bufOffBytes + (unsigned)(Nn * LXS + row * LWS + c4 * 4) * 4u;
        }
#if HAVE_ASYNC
        // fire-and-forget 16B copy from the global address in gsrc to the LDS address in loff; bumps ASYNCcnt
        asm volatile("global_load_async_to_lds_b128 %0, %1, off"
                     :: "v"(loff), "v"(gsrc) : "memory");
        (void)buff;
#else
        *(v4f*)((char*)buff - bufOffBytes + loff) = *(const v4f*)gsrc; // hmm: host/fallback path
#endif
    }
}

__device__ __forceinline__ void wait_async_all() {
#if HAVE_ASYNC
    asm volatile("s_wait_asynccnt 0x0" ::: "memory");
#endif
}

// ---------------------------------------------------------------------------
// Kernel 2: per batch  Y[b] = A2 @ (X[b] @ W^T) + bias   via fp32 WMMA,
// with double-buffered async K-panels.
// ---------------------------------------------------------------------------
__global__ __launch_bounds__(256) void sgc_gemm(
    const float* __restrict__ x,     // (B, 64, 2000)
    const float* __restrict__ W,     // (128, 2000)
    const float* __restrict__ bias,  // (128,)
    const float* __restrict__ A2,    // (64, 64) from workspace
    float* __restrict__ out)         // (B, 64, 128)
{
    __shared__ float sm[2 * PANEL];  // 63488 bytes: two panel buffers
    const unsigned smBase = (unsigned)(unsigned long long)(void*)&sm[0]; // LDS byte offset

    const int tid  = threadIdx.x;
    const int lane = tid & 31;     // wave32
    const int wv   = tid >> 5;     // 0..7
    const int b    = blockIdx.x;
    const float* xb = x + (size_t)b * Nn * Ff;

    v8f acc[4];
    #pragma unroll
    for (int mt = 0; mt < 4; ++mt)
        #pragma unroll
        for (int v = 0; v < 8; ++v) acc[mt][v] = 0.f;

    const int hrow = wv * 16 + (lane & 15);       // this wave's H column
    const int kb   = (lane >> 4) * 2;             // lanes 16-31 hold K+2,K+3

    // prefetch panel 0 into buffer 0
    stage_panel(xb, W, sm, smBase, 0, tid);

    for (int p = 0; p < NP; ++p) {
        const int sel = p & 1;
        float* lx = sm + sel * PANEL;
        float* lw = lx + Nn * LXS;

        wait_async_all();      // this wave's copies for buffer sel are in LDS
        __syncthreads();       // everyone's copies landed; prev buffer free

        if (p + 1 < NP)        // overlap next panel's DMA with this WMMA work
            stage_panel(xb, W, sm + (1 - sel) * PANEL,
                        smBase + (unsigned)(1 - sel) * PANEL * 4u,
                        (p + 1) * KB, tid);

        #pragma unroll 2
        for (int ks = 0; ks < KB; ks += 4) {
            v2f bf;                                   // B frag: W^T 4x16
            bf.x = lw[hrow * LWS + ks + kb];
            bf.y = lw[hrow * LWS + ks + kb + 1];
            #pragma unroll
            for (int mt = 0; mt < 4; ++mt) {
                int mrow = mt * 16 + (lane & 15);
                v2f af;                               // A frag: X 16x4
                af.x = lx[mrow * LXS + ks + kb];
                af.y = lx[mrow * LXS + ks + kb + 1];
                acc[mt] = __builtin_amdgcn_wmma_f32_16x16x4_f32(
                    false, af, false, bf, (short)0, acc[mt], false, false);
            }
        }
    }

    // spill Z = X[b] @ W^T to LDS (C-layout: VGPR v -> M = v + 8*(lane>=16))
    __syncthreads();
    const int ZS = 132;
    float* lz = sm;
    {
        const int ncol  = wv * 16 + (lane & 15);
        const int mhalf = (lane >> 4) * 8;
        #pragma unroll
        for (int mt = 0; mt < 4; ++mt)
            #pragma unroll
            for (int v = 0; v < 8; ++v)
                lz[(mt * 16 + mhalf + v) * ZS + ncol] = acc[mt][v];
    }
    __syncthreads();

    // Y[b] = A2 @ Z + bias  (64x64 @ 64x128; A2 is L2-resident, 16 KB)
    float* outb = out + (size_t)b * Nn * Hh;
    for (int i = tid; i < Nn * Hh; i += 256) {
        int m = i >> 7, h = i & 127;
        float s = bias[h];
        #pragma unroll 8
        for (int n = 0; n < Nn; ++n) s += A2[m * Nn + n] * lz[n * ZS + h];
        outb[i] = s;
    }
}

// ---------------------------------------------------------------------------
extern "C" void kernel_launch(void* const* d_in, const int* in_sizes, int n_in,
                              void* d_out, int out_size, void* d_ws, size_t ws_size,
                              hipStream_t stream) {
    (void)in_sizes; (void)n_in; (void)out_size; (void)ws_size;
    const float*     x    = (const float*)d_in[0];
    const long long* ei   = (const long long*)d_in[1];
    const float*     ew   = (const float*)d_in[2];
    const float*     W    = (const float*)d_in[3];
    const float*     bias = (const float*)d_in[4];
    float* A2  = (float*)d_ws;
    float* out = (float*)d_out;

    hipLaunchKernelGGL(sgc_build_a2, dim3(1), dim3(256), 0, stream, ei, ew, A2);
    hipLaunchKernelGGL(sgc_gemm, dim3(Bb), dim3(256), 0, stream, x, W, bias, A2, out);
}